// gtknn_27341761806801
// MI455X (gfx1250) — compile-verified
//
#include <hip/hip_runtime.h>

typedef __attribute__((ext_vector_type(2))) float v2f;
typedef __attribute__((ext_vector_type(8))) float v8f;

#define BATCH   4
#define CDIM    128
#define NPTS    4096
#define KTOP    16
#define ROWS    128          // query rows per block
#define MCH     32           // database chunk width (double-buffered)
#define NCHUNK  (NPTS / MCH)
#define THREADS 256          // 8 wave32
#define WAVES   8
#define DPAD    (MCH + 4)    // padded D-spill row stride (bank-conflict free)

typedef const __attribute__((address_space(3))) void* lds_cvptr_t;

// 32-bit LDS byte offset of a generic pointer into __shared__ memory
__device__ __forceinline__ unsigned lds_addr_of(const void* p) {
    return (unsigned)(unsigned long long)(lds_cvptr_t)p;
}

// CDNA5 async global->LDS copy, 16B per lane, tracked by ASYNCcnt
__device__ __forceinline__ void async_copy_b128(unsigned lds_byte_addr,
                                                unsigned gvoff_bytes,
                                                const float* gbase) {
    asm volatile("global_load_async_to_lds_b128 %0, %1, %2"
                 :: "v"(lds_byte_addr), "v"(gvoff_bytes), "s"(gbase)
                 : "memory");
}

__device__ __forceinline__ void wait_async0() {
    asm volatile("s_wait_asynccnt 0x0" ::: "memory");
}

// One pass: for each column n of X (C x NPTS, c-major), find the k smallest
// Euclidean distances to columns of Y. Fused WMMA-f32 GEMM + streaming top-16.
__global__ __launch_bounds__(THREADS)
void knn_topk_wmma_kernel(const float* __restrict__ X,
                          const float* __restrict__ Y,
                          float* __restrict__ outD,   // (BATCH, KTOP, NPTS)
                          int*   __restrict__ outI)   // (BATCH, KTOP, NPTS)
{
    __shared__ float As[CDIM * ROWS];              // As[c*ROWS + i] = X[c, n0+i]; reused for merge lists
    __shared__ float a2s[ROWS];                    // |X[:,n0+i]|^2
    __shared__ float b2all[NPTS];                  // |Y[:,m]|^2
    __shared__ float Bs0[CDIM * MCH];              // double-buffered B panel
    __shared__ float Bs1[CDIM * MCH];
    __shared__ float Dspill[WAVES * 16 * DPAD];    // per-wave 16 x MCH inner-product spill

    const int tid = threadIdx.x;
    const int wv  = tid >> 5;        // wave id 0..7
    const int ln  = tid & 31;        // lane
    const int r   = ln & 15;         // row-in-tile / col-in-tile
    const int h   = ln >> 4;         // lane half (selects K pair in frags)

    const int blocksPerBatch = NPTS / ROWS;
    const int batch = blockIdx.x / blocksPerBatch;
    const int n0    = (blockIdx.x % blocksPerBatch) * ROWS;

    const float* Xb = X + (size_t)batch * CDIM * NPTS;
    const float* Yb = Y + (size_t)batch * CDIM * NPTS;

    // ---- per-thread async-staging descriptors: 4 x b128 per chunk ----
    // lin -> (c = lin/8, jv = lin%8): element [c][jv*4 .. jv*4+3] of the chunk
    unsigned gvo[4], lo0[4], lo1[4];
    #pragma unroll
    for (int q = 0; q < 4; ++q) {
        int lin = q * THREADS + tid;
        int c   = lin >> 3;
        int jv  = lin & 7;
        gvo[q] = (unsigned)(((size_t)c * NPTS + jv * 4) * sizeof(float));
        lo0[q] = lds_addr_of(&Bs0[c * MCH + jv * 4]);
        lo1[q] = lds_addr_of(&Bs1[c * MCH + jv * 4]);
    }

    // ---- kick off async copy of chunk 0 into Bs0 (overlaps with setup) ----
    #pragma unroll
    for (int q = 0; q < 4; ++q) async_copy_b128(lo0[q], gvo[q], Yb);

    // ---- stage A panel (128 c x 128 n), float4 vectorized ----
    #pragma unroll
    for (int q = 0; q < (CDIM * ROWS / 4) / THREADS; ++q) {
        int lin = q * THREADS + tid;
        int c   = lin >> 5;          // / (ROWS/4)
        int iv  = lin & 31;
        const float4 v = *(const float4*)(Xb + (size_t)c * NPTS + n0 + iv * 4);
        float* dst = &As[c * ROWS + iv * 4];
        dst[0] = v.x; dst[1] = v.y; dst[2] = v.z; dst[3] = v.w;
    }
    // ---- |Y[:,m]|^2 for all m (L2-resident reads) ----
    for (int q = 0; q < NPTS / THREADS; ++q) {
        int m = q * THREADS + tid;
        float s = 0.f;
        for (int c = 0; c < CDIM; ++c) {
            float v = Yb[(size_t)c * NPTS + m];
            s = fmaf(v, v, s);
        }
        b2all[m] = s;
    }
    __syncthreads();
    if (tid < ROWS) {
        float s = 0.f;
        for (int c = 0; c < CDIM; ++c) {
            float v = As[c * ROWS + tid];
            s = fmaf(v, v, s);
        }
        a2s[tid] = s;
    }
    __syncthreads();

    const int   aoff = wv * 16 + r;        // this lane's query row within block
    const float a2r  = a2s[aoff];

    // sorted ascending (d2, idx) top-16 per lane (2 lanes cooperate per row)
    float bd[KTOP];
    int   bi[KTOP];
    #pragma unroll
    for (int i = 0; i < KTOP; ++i) { bd[i] = 3.4e38f; bi[i] = 0x7fffffff; }

    for (int ch = 0; ch < NCHUNK; ++ch) {
        const int m0 = ch * MCH;
        wait_async0();                      // this wave's copy of chunk ch done
        __syncthreads();                    // every wave's copy done; prev reads of other buf done

        // issue async copy of chunk ch+1 into the other buffer
        if (ch + 1 < NCHUNK) {
            const float* gnext = Yb + (size_t)(m0 + MCH);
            const unsigned* lon = (ch & 1) ? lo0 : lo1;
            #pragma unroll
            for (int q = 0; q < 4; ++q) async_copy_b128(lon[q], gvo[q], gnext);
        }

        const float* Bsp = (ch & 1) ? Bs1 : Bs0;

        // ---- WMMA: 16 rows x 32 cols per wave, K=128 in steps of 4 ----
        v8f acc0 = {}, acc1 = {};
        #pragma unroll 8
        for (int kk = 0; kk < CDIM / 4; ++kk) {
            const int cb = 4 * kk + 2 * h;     // lane-half K pair
            v2f af;                            // A 16x4 f32 frag
            af.x = As[cb * ROWS + aoff];
            af.y = As[(cb + 1) * ROWS + aoff];
            v2f bf;                            // B 4x16 f32 frags
            bf.x = Bsp[cb * MCH + r];
            bf.y = Bsp[(cb + 1) * MCH + r];
            acc0 = __builtin_amdgcn_wmma_f32_16x16x4_f32(false, af, false, bf,
                                                         (short)0, acc0, false, false);
            bf.x = Bsp[cb * MCH + 16 + r];
            bf.y = Bsp[(cb + 1) * MCH + 16 + r];
            acc1 = __builtin_amdgcn_wmma_f32_16x16x4_f32(false, af, false, bf,
                                                         (short)0, acc1, false, false);
        }

        // ---- spill wave's 16x32 inner-product tile (wave-private region) ----
        float* Dw = &Dspill[wv * 16 * DPAD];
        #pragma unroll
        for (int v = 0; v < 8; ++v) {
            int row = v + 8 * h;               // C/D layout: lanes16-31 hold M=v+8
            Dw[row * DPAD + r]      = acc0[v];
            Dw[row * DPAD + 16 + r] = acc1[v];
        }
        // wave-local readback: row-parallel top-k scan (lane pair per row)
        const float* Drow = &Dw[r * DPAD];
        #pragma unroll 4
        for (int s = 0; s < MCH / 2; ++s) {
            int   jl = 2 * s + h;
            float ab = Drow[jl];
            float d2 = a2r + b2all[m0 + jl] - 2.0f * ab;
            int   m  = m0 + jl;
            if (d2 < bd[KTOP - 1] || (d2 == bd[KTOP - 1] && m < bi[KTOP - 1])) {
                float cv = d2; int ci = m;     // branch-free sorted insertion
                #pragma unroll
                for (int i = 0; i < KTOP; ++i) {
                    bool  t  = (cv < bd[i]) || (cv == bd[i] && ci < bi[i]);
                    float tv = bd[i]; int ti = bi[i];
                    bd[i] = t ? cv : bd[i];
                    bi[i] = t ? ci : bi[i];
                    cv = t ? tv : cv;
                    ci = t ? ti : ci;
                }
            }
        }
    }

    // ---- merge the two per-row lists and emit sorted k results ----
    __syncthreads();                           // A panel dead: reuse as list storage
    {
        float* listD = As;                                // 4096 floats
        int*   listI = (int*)As + WAVES * 32 * KTOP;      // 4096 ints
        int g = wv * 32 + ln;
        #pragma unroll
        for (int i = 0; i < KTOP; ++i) {
            listD[g * KTOP + i] = bd[i];
            listI[g * KTOP + i] = bi[i];
        }
    }
    __syncthreads();
    if (h == 0) {
        const float* dA = &As[(wv * 32 + r) * KTOP];
        const float* dB = &As[(wv * 32 + 16 + r) * KTOP];
        const int*   iA = (const int*)As + WAVES * 32 * KTOP + (wv * 32 + r) * KTOP;
        const int*   iB = (const int*)As + WAVES * 32 * KTOP + (wv * 32 + 16 + r) * KTOP;
        int ia = 0, ib = 0;
        const int n = n0 + wv * 16 + r;
        for (int kk = 0; kk < KTOP; ++kk) {
            float da = dA[ia], db = dB[ib];
            int   na = iA[ia], nb = iB[ib];
            bool  tA = (da < db) || (da == db && na < nb);
            float d2  = tA ? da : db;
            int   idx = tA ? na : nb;
            ia += tA ? 1 : 0;
            ib += tA ? 0 : 1;
            outD[((size_t)batch * KTOP + kk) * NPTS + n] = sqrtf(fmaxf(d2, 0.f));
            outI[((size_t)batch * KTOP + kk) * NPTS + n] = idx;
        }
    }
}

extern "C" void kernel_launch(void* const* d_in, const int* in_sizes, int n_in,
                              void* d_out, int out_size, void* d_ws, size_t ws_size,
                              hipStream_t stream) {
    const float* a = (const float*)d_in[0];   // (B, C, N) f32
    const float* b = (const float*)d_in[1];   // (B, C, M) f32
    (void)in_sizes; (void)n_in; (void)d_ws; (void)ws_size; (void)out_size;

    const size_t BKN = (size_t)BATCH * KTOP * NPTS;   // 262144 elements
    float* dist1 = (float*)d_out;                     // (B, k, N)
    float* dist2 = (float*)d_out + BKN;               // (B, k, M)
    int*   idx1  = (int*)d_out + 2 * BKN;             // (B, k, N)
    int*   idx2  = (int*)d_out + 3 * BKN;             // (B, k, M)

    dim3 grid(BATCH * (NPTS / ROWS));
    dim3 block(THREADS);
    // pass 1: for each a-point, k nearest b-points
    knn_topk_wmma_kernel<<<grid, block, 0, stream>>>(a, b, dist1, idx1);
    // pass 2: for each b-point, k nearest a-points
    knn_topk_wmma_kernel<<<grid, block, 0, stream>>>(b, a, dist2, idx2);
}